// EvalCriterion_68951404970521
// MI455X (gfx1250) — compile-verified
//
#include <hip/hip_runtime.h>
#include <hip/hip_bf16.h>

// Problem constants (match reference).
#define Wn   30
#define Bn   16
#define Sn   65536
#define Ln   (Sn + Wn - 1)        // 65565 positions per batch
#define POS  256                  // positions per block (= blockDim.x)
#define MAXROWS (POS + Wn - 1)    // 285 staged rows -> halo overhead 1.11x
#define CHUNKS ((Ln + POS - 1) / POS)  // 257

// ---- CDNA5 async global->LDS copy (ASYNCcnt path) -------------------------
typedef int v2i_t __attribute__((vector_size(2 * sizeof(int))));

__device__ __forceinline__ void async_copy_b64(void* lds_dst, const void* gsrc) {
#if __has_builtin(__builtin_amdgcn_global_load_async_to_lds_b64)
  __builtin_amdgcn_global_load_async_to_lds_b64(
      (__attribute__((address_space(1))) v2i_t*)(void*)gsrc,
      (__attribute__((address_space(3))) v2i_t*)lds_dst,
      0, 0);
#else
  unsigned loff = (unsigned)(unsigned long long)
      (__attribute__((address_space(3))) void*)lds_dst;
  asm volatile("global_load_async_to_lds_b64 %0, %1, off"
               :: "v"(loff), "v"(gsrc) : "memory");
#endif
}

__device__ __forceinline__ void async_wait_all() {
#if __has_builtin(__builtin_amdgcn_s_wait_asynccnt)
  __builtin_amdgcn_s_wait_asynccnt(0);
#else
  asm volatile("s_wait_asynccnt 0x0" ::: "memory");
#endif
}

// ---- zero the 4 output counters (harness poisons d_out) -------------------
__global__ void zero_out_kernel(float* __restrict__ out) {
  if (threadIdx.x < 4) out[threadIdx.x] = 0.0f;
}

// ---- main kernel ----------------------------------------------------------
__global__ __launch_bounds__(POS)
void eval_criterion_kernel(const float* __restrict__ input,
                           const float* __restrict__ target,
                           const int*   __restrict__ lens,
                           const float* __restrict__ tptr,
                           float*       __restrict__ out) {
  __shared__ float lds_i[MAXROWS * Wn];   // 34,200 B
  __shared__ float lds_t[MAXROWS * Wn];   // 34,200 B
  __shared__ int   sh[4];

  const int tid = threadIdx.x;
  const int b   = blockIdx.y;
  const int p0  = blockIdx.x * POS;

  const int r0    = (p0 - (Wn - 1)) > 0 ? (p0 - (Wn - 1)) : 0;
  const int r1    = (p0 + POS) < Sn ? (p0 + POS) : Sn;
  const int nrows = r1 - r0;
  const int nch   = nrows * 15;       // 8-byte chunks per staged array (row = 15 chunks)

  if (tid < 4) sh[tid] = 0;

  // Stage the row span through the async DMA path (per-lane gather via lens_index).
  for (int k = tid; k < nch; k += POS) {
    const int row = k / 15;
    const int cp  = k - row * 15;     // which b64 chunk within the 30-float row
    const size_t grow = (size_t)(unsigned)lens[(size_t)b * Sn + r0 + row];
    const float* g_i = input  + grow * Wn + (size_t)cp * 2;
    const float* g_t = target + grow * Wn + (size_t)cp * 2;
    async_copy_b64(&lds_i[k * 2], g_i);
    async_copy_b64(&lds_t[k * 2], g_t);
  }
  async_wait_all();
  __syncthreads();

  const int p = p0 + tid;
  if (p < Ln) {
    float ai = 0.f, ci = 0.f, at = 0.f, ct = 0.f;
    const bool interior = (p0 >= (Wn - 1)) && (p0 + POS <= Sn);
    if (interior) {
#pragma unroll
      for (int o = 0; o < Wn; ++o) {
        const int idx = (p - o - r0) * Wn + o;
        const float vi = lds_i[idx];
        const float vt = lds_t[idx];
        ai += vi; if (vi != 0.f) ci += 1.f;
        at += vt; if (vt != 0.f) ct += 1.f;
      }
    } else {
#pragma unroll
      for (int o = 0; o < Wn; ++o) {
        const int r = p - o;
        if (r >= r0 && r < r1) {
          const int idx = (r - r0) * Wn + o;
          const float vi = lds_i[idx];
          const float vt = lds_t[idx];
          ai += vi; if (vi != 0.f) ci += 1.f;
          at += vt; if (vt != 0.f) ct += 1.f;
        }
      }
    }
    if (ci <= 0.f) ci = 1.f;
    if (ct <= 0.f) ct = 1.f;
    const float av_i = ai / ci;
    const float av_t = at / ct;
    const float sg   = 1.f / (1.f + expf(-av_i));
    const int pred = (sg > tptr[0]) ? 1 : 0;
    const int lab  = (int)truncf(av_t);
    // out order: TP(0), TN(1), FP(2), FN(3)
    const int cat = pred ? (lab ? 0 : 2) : (lab ? 3 : 1);
    atomicAdd(&sh[cat], 1);
  }
  __syncthreads();

  if (tid < 4 && sh[tid] != 0) {
    atomicAdd(&out[tid], (float)sh[tid]);  // counts <= 2^24: exact in f32
  }
}

// ---- host entry -----------------------------------------------------------
extern "C" void kernel_launch(void* const* d_in, const int* in_sizes, int n_in,
                              void* d_out, int out_size, void* d_ws, size_t ws_size,
                              hipStream_t stream) {
  (void)in_sizes; (void)n_in; (void)out_size; (void)d_ws; (void)ws_size;
  const float* input  = (const float*)d_in[0];
  const float* target = (const float*)d_in[1];
  const int*   lens   = (const int*)d_in[2];
  const float* tptr   = (const float*)d_in[3];
  float* out = (float*)d_out;

  zero_out_kernel<<<1, 32, 0, stream>>>(out);
  dim3 grid(CHUNKS, Bn, 1);
  eval_criterion_kernel<<<grid, POS, 0, stream>>>(input, target, lens, tptr, out);
}